// SemiGCon_2740189135112
// MI455X (gfx1250) — compile-verified
//
#include <hip/hip_runtime.h>
#include <hip/hip_bf16.h>

typedef float v2f __attribute__((ext_vector_type(2)));
typedef float v8f __attribute__((ext_vector_type(8)));

// ---------------------------------------------------------------------------
// Small elementwise kernels
// ---------------------------------------------------------------------------
__global__ void k_fill_f32(float* __restrict__ p, float v, int n) {
    int i = blockIdx.x * blockDim.x + threadIdx.x;
    if (i < n) p[i] = v;
}

__global__ void k_deg_accum(const int* __restrict__ dst, float* __restrict__ deg, int E) {
    int i = blockIdx.x * blockDim.x + threadIdx.x;
    if (i < E) atomicAdd(&deg[dst[i]], 1.0f);
}

__global__ void k_rsqrt_inplace(float* __restrict__ p, int n) {
    int i = blockIdx.x * blockDim.x + threadIdx.x;
    if (i < n) p[i] = rsqrtf(p[i]);
}

// out[row][c] = h[row][c] * dinv[row]^2 + bias[c]   (self-loop term + bias)
template <int C>
__global__ void k_selfloop_bias(const float* __restrict__ h, const float* __restrict__ dinv,
                                const float* __restrict__ bias, float* __restrict__ out,
                                long total) {
    long gid = (long)blockIdx.x * blockDim.x + threadIdx.x;
    if (gid >= total) return;
    constexpr int SH = (C == 64) ? 6 : 5;
    int  c   = (int)(gid & (C - 1));
    long row = gid >> SH;
    float di = dinv[row];
    out[gid] = h[gid] * di * di + bias[c];
}

// one thread per (edge, channel): out[dst][c] += h[src][c] * dinv[src]*dinv[dst]
// A wave covers 32 consecutive channels of one edge -> index/dinv loads are
// wave-uniform (single request), h loads and atomics are fully coalesced.
template <int C>
__global__ void k_scatter_add(const int* __restrict__ src, const int* __restrict__ dst,
                              const float* __restrict__ dinv, const float* __restrict__ h,
                              float* __restrict__ out, long total) {
    long gid = (long)blockIdx.x * blockDim.x + threadIdx.x;
    if (gid >= total) return;
    constexpr int SH = (C == 64) ? 6 : 5;
    int  c = (int)(gid & (C - 1));
    long e = gid >> SH;
    int  s = src[e];
    int  d = dst[e];
    float w = dinv[s] * dinv[d];
    atomicAdd(&out[(long)d * C + c], h[(long)s * C + c] * w);
}

// ---------------------------------------------------------------------------
// WMMA f32 GEMM:  H[n x COLS] = act(X[n x K]) @ W[K x COLS]
// One wave computes a 16-row strip across all COLS columns using
// V_WMMA_F32_16X16X4_F32 (full fp32 — this problem is bandwidth bound,
// the f32 matrix pipe is free and keeps reference precision).
// 8 waves / block; W staged in LDS once per block.
//
// Layouts (ISA 7.12.2):
//   A 16x4 : lane%16 = M ; lanes 0-15 hold K=k0+{0,1}, lanes 16-31 K=k0+{2,3}
//   B 4x16 : lane%16 = N ; same half-wave K split (mirror of A)
//   C 16x16: VGPR i -> M=i (lanes 0-15), M=i+8 (lanes 16-31), N=lane%16
// ---------------------------------------------------------------------------
template <int K, int COLS, bool RELU>
__global__ void k_gemm_wmma(const float* __restrict__ X, const float* __restrict__ W,
                            float* __restrict__ H, int nStrips, int nRows) {
    __shared__ float lW[K * COLS];
    for (int i = threadIdx.x; i < K * COLS; i += blockDim.x) lW[i] = W[i];
    __syncthreads();

    int wave  = threadIdx.x >> 5;
    int lane  = threadIdx.x & 31;
    int strip = blockIdx.x * 8 + wave;
    if (strip >= nStrips) return;           // wave-uniform: EXEC stays all-ones

    int m0   = strip << 4;
    int half = lane >> 4;                   // 0 or 1
    int lm   = lane & 15;

    int arow = m0 + lm;
    if (arow >= nRows) arow = nRows - 1;    // clamp keeps EXEC full for WMMA
    const float* xrow = X + (size_t)arow * K;

    constexpr int NT = COLS / 16;
    v8f acc[NT];
#pragma unroll
    for (int t = 0; t < NT; ++t) acc[t] = (v8f){0.f, 0.f, 0.f, 0.f, 0.f, 0.f, 0.f, 0.f};

    for (int k0 = 0; k0 < K; k0 += 4) {
        int kl = k0 + half * 2;
        float a0 = xrow[kl];
        float a1 = xrow[kl + 1];
        if (RELU) { a0 = fmaxf(a0, 0.f); a1 = fmaxf(a1, 0.f); }
        v2f a = {a0, a1};
#pragma unroll
        for (int t = 0; t < NT; ++t) {
            int col = t * 16 + lm;
            v2f b = {lW[kl * COLS + col], lW[(kl + 1) * COLS + col]};
            acc[t] = __builtin_amdgcn_wmma_f32_16x16x4_f32(
                false, a, false, b, (short)0, acc[t], false, false);
        }
    }

    // Epilogue: wave-uniform fast path for full strips (the common case:
    // nRows % 16 == 0) -> straight unguarded coalesced stores, no EXEC churn.
    float* hb = H + (size_t)(m0 + half * 8) * COLS + lm;
    if (m0 + 16 <= nRows) {
#pragma unroll
        for (int t = 0; t < NT; ++t)
#pragma unroll
            for (int i = 0; i < 8; ++i)
                hb[(size_t)i * COLS + t * 16] = acc[t][i];
    } else {
#pragma unroll
        for (int t = 0; t < NT; ++t)
#pragma unroll
            for (int i = 0; i < 8; ++i)
                if (m0 + half * 8 + i < nRows)
                    hb[(size_t)i * COLS + t * 16] = acc[t][i];
    }
}

// ---------------------------------------------------------------------------
// Column stats (32 cols): double accumulation; LDS tree over 8 row-groups,
// then one f64 atomic per column per block.  stats[0..31]=sum, [32..63]=sumsq
// ---------------------------------------------------------------------------
__global__ void k_zero_stats(double* __restrict__ s) {
    int i = threadIdx.x;
    if (i < 64) s[i] = 0.0;
}

__global__ void k_colstats(const float* __restrict__ o, double* __restrict__ stats, int n) {
    __shared__ double ls[256];
    __shared__ double ls2[256];
    int c = threadIdx.x & 31;
    int g = threadIdx.x >> 5;
    double s = 0.0, s2 = 0.0;
    for (int row = blockIdx.x * 8 + g; row < n; row += gridDim.x * 8) {
        double v = (double)o[(size_t)row * 32 + c];
        s  += v;
        s2 += v * v;
    }
    ls[threadIdx.x]  = s;
    ls2[threadIdx.x] = s2;
    __syncthreads();
    for (int off = 128; off >= 32; off >>= 1) {
        if (threadIdx.x < off) {
            ls[threadIdx.x]  += ls[threadIdx.x + off];
            ls2[threadIdx.x] += ls2[threadIdx.x + off];
        }
        __syncthreads();
    }
    if (threadIdx.x < 32) {
        atomicAdd(&stats[c],      ls[threadIdx.x]);
        atomicAdd(&stats[32 + c], ls2[threadIdx.x]);
    }
}

__global__ void k_znorm(float* __restrict__ o, const double* __restrict__ stats, int n) {
    long gid = (long)blockIdx.x * blockDim.x + threadIdx.x;
    long total = (long)n * 32;
    if (gid >= total) return;
    int c = (int)(gid & 31);
    double sum = stats[c];
    double sq  = stats[32 + c];
    double mean = sum / (double)n;
    double var  = (sq - sum * sum / (double)n) / (double)(n - 1);  // unbiased (ddof=1)
    float  inv  = (float)(1.0 / sqrt(var));
    o[gid] = (o[gid] - (float)mean) * inv;
}

// ---------------------------------------------------------------------------
// Host orchestration
// ---------------------------------------------------------------------------
extern "C" void kernel_launch(void* const* d_in, const int* in_sizes, int n_in,
                              void* d_out, int out_size, void* d_ws, size_t ws_size,
                              hipStream_t stream) {
    const int IN_DIM = 128, HID = 64, OUT = 32;
    const int n = in_sizes[0] / IN_DIM;   // 50000
    const int E = in_sizes[1] / 2;        // 800000

    const float* W1 = (const float*)d_in[4];
    const float* b1 = (const float*)d_in[5];
    const float* W2 = (const float*)d_in[6];
    const float* b2 = (const float*)d_in[7];

    // workspace carve-out (256B aligned)
    char* ws = (char*)d_ws;
    size_t off = 0;
    auto carve = [&](size_t bytes) -> char* {
        char* p = ws + off;
        off += (bytes + 255) & ~(size_t)255;
        return p;
    };
    float*  dinv  = (float*)carve((size_t)n * sizeof(float));
    float*  h1    = (float*)carve((size_t)n * HID * sizeof(float));
    float*  agg   = (float*)carve((size_t)n * HID * sizeof(float));
    float*  h2    = (float*)carve((size_t)n * OUT * sizeof(float));
    double* stats = (double*)carve(64 * sizeof(double));
    (void)ws_size;

    const int strips  = (n + 15) / 16;
    const int gblocks = (strips + 7) / 8;

    for (int g = 0; g < 2; ++g) {
        const float* x   = (const float*)d_in[g * 2 + 0];
        const int*   ei  = (const int*)  d_in[g * 2 + 1];
        const int*   src = ei;
        const int*   dst = ei + E;
        float* outg = (float*)d_out + (size_t)g * n * OUT;

        // symmetric normalization: deg over dst (incl. self-loop) -> dinv
        k_fill_f32<<<(n + 255) / 256, 256, 0, stream>>>(dinv, 1.0f, n);
        k_deg_accum<<<(E + 255) / 256, 256, 0, stream>>>(dst, dinv, E);
        k_rsqrt_inplace<<<(n + 255) / 256, 256, 0, stream>>>(dinv, n);

        // layer 1: h1 = x @ W1 ; agg = selfloop + bias ; scatter edges
        k_gemm_wmma<128, 64, false><<<gblocks, 256, 0, stream>>>(x, W1, h1, strips, n);
        {
            long t = (long)n * HID;
            k_selfloop_bias<64><<<(int)((t + 255) / 256), 256, 0, stream>>>(h1, dinv, b1, agg, t);
            long te = (long)E * HID;
            k_scatter_add<64><<<(int)((te + 255) / 256), 256, 0, stream>>>(src, dst, dinv, h1, agg, te);
        }

        // layer 2: h2 = relu(agg) @ W2 ; out = selfloop + bias ; scatter edges
        k_gemm_wmma<64, 32, true><<<gblocks, 256, 0, stream>>>(agg, W2, h2, strips, n);
        {
            long t = (long)n * OUT;
            k_selfloop_bias<32><<<(int)((t + 255) / 256), 256, 0, stream>>>(h2, dinv, b2, outg, t);
            long te = (long)E * OUT;
            k_scatter_add<32><<<(int)((te + 255) / 256), 256, 0, stream>>>(src, dst, dinv, h2, outg, te);
        }

        // column z-score (unbiased std)
        k_zero_stats<<<1, 64, 0, stream>>>(stats);
        k_colstats<<<512, 256, 0, stream>>>(outg, stats, n);
        {
            long t = (long)n * OUT;
            k_znorm<<<(int)((t + 255) / 256), 256, 0, stream>>>(outg, stats, n);
        }
    }
}